// Generator_42142219108429
// MI455X (gfx1250) — compile-verified
//
#include <hip/hip_runtime.h>
#include <math.h>

// ---------------------------------------------------------------------------
// Pipeline constants (reference: C=128, H_FEAT=256, STYLE=512, K=3,
// flow 512x512, cut=1 -> s=254, p=254/4+254/8=94, padded=442)
// ---------------------------------------------------------------------------
#define PD   94
#define SC   254
#define PS   442
#define NPS  (442*442)
#define HF   256
#define CCH  128
#define FLOWS 512
#define PPS  258            // padded activation row (256 + halo)
#define PA   (258*258)      // padded activation plane (per channel-pair, x2 floats)

typedef __attribute__((ext_vector_type(2))) float v2f;
typedef __attribute__((ext_vector_type(4))) float v4f;
typedef __attribute__((ext_vector_type(8))) float v8f;

__device__ __forceinline__ int reflect254(int j) {
  if (j < 0)  j = -j;             // p=94 < 254, one reflection suffices
  if (j >= SC) j = 2*SC - 2 - j;
  return j;
}

// ---------------------------------------------------------------------------
// K1: bilinear-resize flow 512->256 (*0.5), crop [1:255], reflect-pad to 442
// pflow layout: [2][442][442]
// ---------------------------------------------------------------------------
__global__ void k_pflow(const float* __restrict__ flow, float* __restrict__ pflow) {
  int t = blockIdx.x * blockDim.x + threadIdx.x;
  if (t >= 2*NPS) return;
  int ch = t / NPS;
  int rem = t - ch*NPS;
  int y = rem / PS, x = rem - (rem / PS) * PS;
  int ry = reflect254(y - PD) + 1;   // index into 256 grid (after crop 1:-1)
  int rx = reflect254(x - PD) + 1;
  float sy = 2.0f * (float)ry + 0.5f;   // half-pixel mapping 256 -> 512
  float sx = 2.0f * (float)rx + 0.5f;
  int y0 = (int)floorf(sy), x0 = (int)floorf(sx);
  float fy = sy - (float)y0, fx = sx - (float)x0;
  int y1 = min(y0 + 1, FLOWS - 1), x1 = min(x0 + 1, FLOWS - 1);
  y0 = min(max(y0, 0), FLOWS - 1);  x0 = min(max(x0, 0), FLOWS - 1);
  const float* f = flow + (size_t)ch * FLOWS * FLOWS;
  float v00 = f[y0*FLOWS + x0], v10 = f[y0*FLOWS + x1];
  float v01 = f[y1*FLOWS + x0], v11 = f[y1*FLOWS + x1];
  float v = (1.0f-fy)*((1.0f-fx)*v00 + fx*v10) + fy*((1.0f-fx)*v01 + fx*v11);
  pflow[t] = 0.5f * v;   // * scale (=256/512)
}

// ---------------------------------------------------------------------------
// K2: Euler integration; per-pixel independent trajectory.
// mode 0: future (steps=idx, +flow); mode 1: past (steps=nf-idx-1, -flow).
// Stores final TARGET coordinates (coord + accumulated flow) -> splat gx,gy.
// ---------------------------------------------------------------------------
__global__ void k_euler(const float* __restrict__ pflow, const int* __restrict__ idxp,
                        const int* __restrict__ nfp, int mode, float* __restrict__ tgt) {
  int t = blockIdx.x * blockDim.x + threadIdx.x;
  if (t >= NPS) return;
  int x = t % PS, y = t / PS;
  int idx = idxp[0], nf = nfp[0];
  int steps = mode ? (nf - idx - 1) : idx;
  float sign = mode ? -1.0f : 1.0f;
  float cx = (float)x, cy = (float)y;
  float dx = cx, dy = cy;
  bool invalid = false;
  for (int s = 0; s < steps; ++s) {
    int rx = (int)rintf(dx); rx = min(max(rx, 0), PS - 1);   // jnp.round = RNE
    int ry = (int)rintf(dy); ry = min(max(ry, 0), PS - 1);
    float fx = sign * pflow[ry*PS + rx];
    float fy = sign * pflow[NPS + ry*PS + rx];
    float nx = dx + fx, ny = dy + fy;
    invalid = invalid || (nx > (float)(PS-1)) || (nx < 0.0f) ||
                         (ny > (float)(PS-1)) || (ny < 0.0f);
    if (invalid) { dx = cx; dy = cy; } else { dx = nx; dy = ny; }
  }
  tgt[t] = dx;
  tgt[NPS + t] = dy;
}

// ---------------------------------------------------------------------------
// K3: zero fill
// ---------------------------------------------------------------------------
__global__ void k_zero(float* __restrict__ p, size_t n) {
  size_t t = (size_t)blockIdx.x * blockDim.x + threadIdx.x;
  if (t < n) p[t] = 0.0f;
}

// ---------------------------------------------------------------------------
// K4: forward bilinear splat (softsplat_sum). Accumulator: [129][442][442].
// Splat weights are constant per half (Zp==1): (1-alpha) / alpha.
// Targets in [0,441]; corners at 442 are exactly what the final crop drops.
// Unconditional atomics (weight 0 for invalid corners) -> no exec branches.
// ---------------------------------------------------------------------------
__global__ void k_splat(const float* __restrict__ feature, const float* __restrict__ tgtf,
                        const float* __restrict__ tgtp, const int* __restrict__ idxp,
                        const int* __restrict__ nfp, float* __restrict__ accum) {
  int t = blockIdx.x * blockDim.x + threadIdx.x;
  if (t >= 2*NPS) return;
  int half = t / NPS;
  int pix  = t - half*NPS;
  int x = pix % PS, y = pix / PS;
  float alpha = (float)idxp[0] / (float)(nfp[0] - 1);
  float wz = half ? alpha : (1.0f - alpha);
  const float* tg = half ? tgtp : tgtf;
  float gx = tg[pix], gy = tg[NPS + pix];
  float x0f = floorf(gx), y0f = floorf(gy);
  float wx1 = gx - x0f, wy1 = gy - y0f;
  int x0 = (int)x0f, y0 = (int)y0f;
  const int dxs[4] = {0,1,0,1};
  const int dys[4] = {0,0,1,1};
  float wgt[4] = {(1.0f-wx1)*(1.0f-wy1), wx1*(1.0f-wy1), (1.0f-wx1)*wy1, wx1*wy1};
  int   ci[4]; float cw[4];
#pragma unroll
  for (int q = 0; q < 4; ++q) {
    int xi = x0 + dxs[q], yi = y0 + dys[q];
    bool valid = (xi >= 0) & (xi < PS) & (yi >= 0) & (yi < PS);
    ci[q] = valid ? (yi*PS + xi) : 0;
    cw[q] = valid ? (wz * wgt[q]) : 0.0f;
  }
  // source value = reflect-padded cropped feature (recomputed, never materialized)
  int ry = reflect254(y - PD) + 1;
  int rx = reflect254(x - PD) + 1;
  const float* fb = feature + (size_t)ry * HF + rx;
  for (int c = 0; c < CCH; ++c) {
    float v = fb[(size_t)c * HF * HF];
    float* ab = accum + (size_t)c * NPS;
#pragma unroll
    for (int q = 0; q < 4; ++q) atomicAdd(ab + ci[q], v * cw[q]);
  }
  float* mb = accum + (size_t)CCH * NPS;
#pragma unroll
  for (int q = 0; q < 4; ++q) atomicAdd(mb + ci[q], cw[q]);
}

// ---------------------------------------------------------------------------
// K5: normalize by metric channel, crop 442 -> 256 (st=93), and scatter into
// the zero-padded, channel-pair-interleaved activation buffer:
//   Xp[ pair ][ y+1 ][ x+1 ][ 2 ]   (pair = c/2, last dim = c&1)
// so the conv A-operand is a single unconditional b64 load per lane.
// ---------------------------------------------------------------------------
__global__ void k_norm(const float* __restrict__ accum, float* __restrict__ Xp) {
  int t = blockIdx.x * blockDim.x + threadIdx.x;
  if (t >= CCH * HF * HF) return;
  int c = t >> 16;
  int pix = t & 65535;
  int y = pix >> 8, x = pix & 255;
  int src = (y + 93) * PS + (x + 93);
  float m = accum[(size_t)CCH * NPS + src];
  if (m == 0.0f) m = 1.0f;
  float v = accum[(size_t)c * NPS + src] / m;
  Xp[(((size_t)(c >> 1)) * PA + (size_t)(y + 1) * PPS + (x + 1)) * 2 + (c & 1)] = v;
}

// ---------------------------------------------------------------------------
// K6: style modulation vector s[128]
// ---------------------------------------------------------------------------
__global__ void k_style(const float* __restrict__ style, const float* __restrict__ mw,
                        const float* __restrict__ mb, float* __restrict__ s) {
  int i = threadIdx.x;
  if (i >= CCH) return;
  float acc = 0.0f;
  const float* row = mw + (size_t)i * 512;
  for (int d = 0; d < 512; ++d) acc += style[d] * row[d];
  s[i] = acc * 0.04419417382415922f /* 1/sqrt(512) */ + mb[i];
}

// ---------------------------------------------------------------------------
// K7: modulate + demodulate weights; layout wTp[tap][pair][out][2] so the
// conv B-operand is one b64 load, contiguous across the 16 out-channel lanes.
// ---------------------------------------------------------------------------
__global__ void k_modw(const float* __restrict__ cw, const float* __restrict__ s,
                       float* __restrict__ wTp) {
  int o = threadIdx.x;
  if (o >= CCH) return;
  const float scale = 0.029462782549439483f;  // 1/sqrt(128*3*3)
  const float* base = cw + (size_t)o * CCH * 9;
  float sum = 0.0f;
  for (int i = 0; i < CCH; ++i) {
    float si = s[i];
    for (int k = 0; k < 9; ++k) { float w = scale * base[i*9 + k] * si; sum += w * w; }
  }
  float demod = rsqrtf(sum + 1e-8f);
  for (int i = 0; i < CCH; ++i) {
    float si = s[i] * scale * demod;
    for (int k = 0; k < 9; ++k)
      wTp[(((size_t)k * 64 + (i >> 1)) * CCH + o) * 2 + (i & 1)] = base[i*9 + k] * si;
  }
}

// ---------------------------------------------------------------------------
// K8: modulated 3x3 conv as implicit GEMM on V_WMMA_F32_16X16X4_F32.
// Block = 8 waves, all on the SAME 16-out-channel tile, 8 consecutive 16-pixel
// tiles -> B panel staged per-tap in LDS (8 KB) and shared by every wave.
// A 16x4 f32: lanes 0-15 K={k,k+1}, lanes 16-31 K={k+2,k+3}  -> one b64 load.
// B 4x16:     VGPR0 rows K=k/K=k+2, VGPR1 rows K=k+1/K=k+3   -> one ds b64.
// k-loop is software-pipelined (prefetch next a/b before the WMMA).
// Epilogue fuses noise, bias, leaky_relu(0.2)*sqrt(2); two b128 stores/lane.
// ---------------------------------------------------------------------------
__global__ __launch_bounds__(256) void k_conv(const float* __restrict__ Xp,
    const float* __restrict__ wTp, const float* __restrict__ bias,
    const float* __restrict__ noise, const float* __restrict__ nstrength,
    float* __restrict__ out) {
  __shared__ float smem[64 * 16 * 2];   // one tap: 64 pairs x 16 outc x 2 = 8 KB
  int tid  = threadIdx.x;
  int lane = tid & 31;
  int wave = tid >> 5;
  int ntile  = blockIdx.x & 7;
  int mgroup = blockIdx.x >> 3;
  int nt16 = ntile << 4;

  int mtile = (mgroup << 3) + wave;
  int pixbase = mtile << 4;
  int y  = pixbase >> 8;
  int xb = pixbase & 255;
  int lx = lane & 15;
  int ph = lane >> 4;                 // 0: pairs {k0/2}, 1: pairs {k0/2+1}
  int n = nt16 + lx;

  const v2f* X2  = (const v2f*)Xp;
  const v2f* wt2 = (const v2f*)wTp;
  v2f* sm2 = (v2f*)smem;

  v8f acc = {0.0f, 0.0f, 0.0f, 0.0f, 0.0f, 0.0f, 0.0f, 0.0f};

  for (int tap = 0; tap < 9; ++tap) {
    int ky = tap / 3, kx = tap - 3 * (tap / 3);
    __syncthreads();                  // previous tap's smem reads done
#pragma unroll
    for (int i = 0; i < 4; ++i) {     // 1024 v2f, 256 threads -> 4 each
      int e = tid + i * 256;
      sm2[e] = wt2[((size_t)tap * 64 + (e >> 4)) * CCH + nt16 + (e & 15)];
    }
    __syncthreads();

    const v2f* ap = X2 + (size_t)ph * PA + (size_t)(y + ky) * PPS + (xb + lx + kx);
    const v2f* bp = sm2 + (ph << 4) + lx;
    __builtin_prefetch((const void*)(ap + (size_t)16 * PA), 0, 3);  // global_prefetch_b8

    v2f a = ap[0];
    v2f b = bp[0];
#pragma unroll 4
    for (int s = 1; s < 32; ++s) {    // channel pairs advance by 2 per step
      v2f an = ap[(size_t)(2 * s) * PA];
      v2f bn = bp[(2 * s) << 4];
      acc = __builtin_amdgcn_wmma_f32_16x16x4_f32(
          false, a, false, b, (short)0, acc, false, false);
      a = an; b = bn;
    }
    acc = __builtin_amdgcn_wmma_f32_16x16x4_f32(
        false, a, false, b, (short)0, acc, false, false);
  }

  // Epilogue: C layout -> lane covers 8 consecutive pixels of one channel.
  float ns = nstrength[0];
  float bi = bias[n];
  int prow = pixbase + ((lane < 16) ? 0 : 8);
  const v4f* np = (const v4f*)(noise + prow);       // 32B aligned (prow % 8 == 0)
  v4f nz0 = np[0], nz1 = np[1];
  v4f r0, r1;
#pragma unroll
  for (int r = 0; r < 4; ++r) {
    float v = acc[r] + ns * nz0[r] + bi;
    r0[r] = (v > 0.0f ? v : 0.2f * v) * 1.4142135623730951f;
  }
#pragma unroll
  for (int r = 0; r < 4; ++r) {
    float v = acc[r + 4] + ns * nz1[r] + bi;
    r1[r] = (v > 0.0f ? v : 0.2f * v) * 1.4142135623730951f;
  }
  v4f* op = (v4f*)(out + (size_t)n * 65536 + prow);
  op[0] = r0;
  op[1] = r1;
}

// ---------------------------------------------------------------------------
// Launch
// ---------------------------------------------------------------------------
extern "C" void kernel_launch(void* const* d_in, const int* in_sizes, int n_in,
                              void* d_out, int out_size, void* d_ws, size_t ws_size,
                              hipStream_t stream) {
  (void)in_sizes; (void)n_in; (void)out_size; (void)ws_size;
  const float* feature = (const float*)d_in[0];   // 1x128x256x256
  const float* flow    = (const float*)d_in[1];   // 1x2x512x512
  const float* style   = (const float*)d_in[2];   // 1x512
  const float* mw      = (const float*)d_in[3];   // 128x512
  const float* mb      = (const float*)d_in[4];   // 128
  const float* cw      = (const float*)d_in[5];   // 128x128x3x3
  const float* cb      = (const float*)d_in[6];   // 128
  const float* noise   = (const float*)d_in[7];   // 1x1x256x256
  const float* nstr    = (const float*)d_in[8];   // 1
  const int*   idxp    = (const int*)d_in[9];     // scalar 30
  const int*   nfp     = (const int*)d_in[10];    // scalar 60
  float* out = (float*)d_out;

  // Workspace layout (floats): ~143 MB total
  float* ws = (float*)d_ws;
  size_t off = 0;
  float* pflow  = ws + off; off += 2  * (size_t)NPS;        // padded flow
  float* tgtf   = ws + off; off += 2  * (size_t)NPS;        // future targets
  float* tgtp   = ws + off; off += 2  * (size_t)NPS;        // past targets
  float* accum  = ws + off; off += 129 * (size_t)NPS;       // splat accumulator
  float* smod   = ws + off; off += 128;                     // style vector
  float* wTp    = ws + off; off += 9 * 64 * 128 * 2;        // weights [tap][pair][o][2]
  float* Xp     = ws + off; off += (size_t)64 * PA * 2;     // padded activations

  const int B = 256;
  k_pflow<<<(2*NPS + B - 1)/B, B, 0, stream>>>(flow, pflow);
  k_euler<<<(NPS + B - 1)/B, B, 0, stream>>>(pflow, idxp, nfp, 0, tgtf);
  k_euler<<<(NPS + B - 1)/B, B, 0, stream>>>(pflow, idxp, nfp, 1, tgtp);
  size_t azn = 129 * (size_t)NPS;
  k_zero<<<(unsigned)((azn + B - 1)/B), B, 0, stream>>>(accum, azn);
  size_t xpn = (size_t)64 * PA * 2;
  k_zero<<<(unsigned)((xpn + B - 1)/B), B, 0, stream>>>(Xp, xpn);
  k_splat<<<(2*NPS + B - 1)/B, B, 0, stream>>>(feature, tgtf, tgtp, idxp, nfp, accum);
  k_norm<<<(CCH*HF*HF + B - 1)/B, B, 0, stream>>>(accum, Xp);
  k_style<<<1, 128, 0, stream>>>(style, mw, mb, smod);
  k_modw<<<1, 128, 0, stream>>>(cw, smod, wTp);
  k_conv<<<4096, 256, 0, stream>>>(Xp, wTp, cb, noise, nstr, out);
}